// MedianBlur_53712861004455
// MI455X (gfx1250) — compile-verified
//
#include <hip/hip_runtime.h>
#include <stdint.h>

// Median blur 7x7, zero-padded, lower median (== exact middle, 49 odd).
// B=8, C=3, H=W=512, fp32.
//
// Strategy (MI455X / gfx1250):
//  - Stage a (32+6)x(8+6) halo tile into LDS using the CDNA5 async
//    global->LDS DMA path (global_load_async_to_lds_b32, ASYNCcnt,
//    s_wait_asynccnt). Zero padding: pre-zero LDS, async-load only valid lanes.
//  - Per pixel: forgetful selection over the 49-element window.
//    Working set of 26 registers, 23 stream-in cycles, 621 compare-exchanges,
//    final v_med3. This is the minimal-state exact-median algorithm; the
//    problem is VALU-bound (order statistics cannot use WMMA).

#define IMG_H 512
#define IMG_W 512
#define NPLANES 24        // 8 * 3
#define TW 32
#define TH 8
#define PAD 3
#define HW (TW + 2 * PAD) // 38
#define HH (TH + 2 * PAD) // 14
#define HALO (HW * HH)    // 532

__device__ __forceinline__ unsigned lds_byte_off(const void* p) {
  // generic LDS pointer: low 32 bits are the LDS byte address (ISA 10.2)
  return (unsigned)(unsigned long long)(uintptr_t)p;
}

__device__ __forceinline__ void async_copy_b32(const float* gsrc, float* ldst) {
  unsigned l = lds_byte_off(ldst);
  // vdst = LDS byte address, vaddr = 64-bit global address, no SADDR
  asm volatile("global_load_async_to_lds_b32 %0, %1, off"
               :
               : "v"(l), "v"(gsrc)
               : "memory");
}

__device__ __forceinline__ void wait_async_zero() {
  asm volatile("s_wait_asynccnt 0" ::: "memory");
}

__device__ __forceinline__ void cas(float& a, float& b) {
  float lo = fminf(a, b);
  float hi = fmaxf(a, b);
  a = lo;
  b = hi;
}

#define WIN(n) smem[(ty + ((n) / 7)) * HW + (tx + ((n) % 7))]

__global__ __launch_bounds__(256) void MedianBlur7_kernel(
    const float* __restrict__ in, float* __restrict__ out) {
  __shared__ float smem[HALO];

  const int tx = threadIdx.x;          // 0..31
  const int ty = threadIdx.y;          // 0..7
  const int tileX = blockIdx.x * TW;
  const int tileY = blockIdx.y * TH;
  const int plane = blockIdx.z;        // b*C + c

  const float* __restrict__ src = in + (size_t)plane * IMG_H * IMG_W;
  float* __restrict__ dst = out + (size_t)plane * IMG_H * IMG_W;

  const int lin = ty * TW + tx;        // 0..255

  // --- 1) zero-fill the LDS halo tile (zero padding semantics) ---
#pragma unroll
  for (int i = 0; i < 3; ++i) {
    int e = lin + i * 256;
    if (e < HALO) smem[e] = 0.0f;
  }
  __syncthreads();  // ds stores committed before async writes can land

  // --- 2) async global->LDS load of the valid sub-rectangle ---
#pragma unroll
  for (int i = 0; i < 3; ++i) {
    int e = lin + i * 256;
    if (e < HALO) {
      int ly = e / HW;
      int lx = e - ly * HW;
      int gy = tileY + ly - PAD;
      int gx = tileX + lx - PAD;
      if (gx >= 0 && gx < IMG_W && gy >= 0 && gy < IMG_H) {
        async_copy_b32(src + (size_t)gy * IMG_W + gx, &smem[e]);
      }
    }
  }
  wait_async_zero();   // this wave's async writes have reached LDS
  __syncthreads();     // all waves' tiles visible

  // --- 3) forgetful selection: exact lower median of 49 ---
  float v[26];
#pragma unroll
  for (int n = 0; n < 26; ++n) v[n] = WIN(n);

#pragma unroll
  for (int c = 0; c < 23; ++c) {
    // active set is v[c..25]; push min to v[c]
#pragma unroll
    for (int i = c + 1; i < 26; ++i) cas(v[c], v[i]);
    // push max of remaining to v[25]
#pragma unroll
    for (int i = c + 1; i < 25; ++i) cas(v[i], v[25]);
    // discard min (v[c], never touched again) and max (overwrite v[25])
    v[25] = WIN(26 + c);
  }
  // remaining active set: v[23..25], target rank 1 of 3 -> median of 3
#if __has_builtin(__builtin_amdgcn_fmed3f)
  float med = __builtin_amdgcn_fmed3f(v[23], v[24], v[25]);
#else
  float mn = fminf(v[23], v[24]);
  float mx = fmaxf(v[23], v[24]);
  float med = fmaxf(mn, fminf(mx, v[25]));
#endif

  dst[(size_t)(tileY + ty) * IMG_W + (tileX + tx)] = med;
}

extern "C" void kernel_launch(void* const* d_in, const int* in_sizes, int n_in,
                              void* d_out, int out_size, void* d_ws,
                              size_t ws_size, hipStream_t stream) {
  (void)in_sizes; (void)n_in; (void)out_size; (void)d_ws; (void)ws_size;
  const float* img = (const float*)d_in[0];
  float* out = (float*)d_out;

  dim3 block(TW, TH, 1);                                   // 256 = 8 waves
  dim3 grid(IMG_W / TW, IMG_H / TH, NPLANES);              // 16 x 64 x 24
  MedianBlur7_kernel<<<grid, block, 0, stream>>>(img, out);
}